// ODEFunc_20710332302010
// MI455X (gfx1250) — compile-verified
//
#include <hip/hip_runtime.h>
#include <cstdint>
#include <cstddef>

// Problem geometry (fixed by the reference setup).
#define B_       64
#define H_       1024
#define W_       1024
#define TILE     32          // rows swept per block
#define NTHREADS 256         // W_/4 lanes, one float4 chunk per lane per row

static_assert(W_ / 4 == NTHREADS, "one b128 chunk per thread per row");

// Payload type the async-LDS builtin expects: 16-byte int vector.
typedef int v4i __attribute__((vector_size(16)));
typedef __attribute__((address_space(1))) v4i gv4i;  // global (device) AS
typedef __attribute__((address_space(3))) v4i lv4i;  // LDS AS

// One 16-byte async copy: global -> LDS, tracked by ASYNCcnt, no VGPR data path.
__device__ __forceinline__ void async_b128(const float* g, float* l) {
  __builtin_amdgcn_global_load_async_to_lds_b128(
      (gv4i*)(void*)g, (lv4i*)(void*)l, /*offset=*/0, /*cpol=*/0);
}

__device__ __forceinline__ void wait_async0() {
  asm volatile("s_wait_asynccnt 0x0" ::: "memory");
}

__global__ __launch_bounds__(NTHREADS)
void ODEFunc_20710332302010_kernel(const float* __restrict__ y,
                                   const float* __restrict__ zX,
                                   const float* __restrict__ zY,
                                   const float* __restrict__ zR,
                                   const int*   __restrict__ dxp,
                                   const int*   __restrict__ dyp,
                                   float*       __restrict__ out) {
  __shared__ __align__(16) float sm[4][W_];   // 4-slot row ring buffer (16 KB)

  const int tid = threadIdx.x;
  const int x4  = tid * 4;                    // first column this lane owns
  const int b   = blockIdx.y;
  const int h0  = blockIdx.x * TILE;

  // Per-batch scalars (uniform -> s_load path), dx/dy read from device memory.
  const float azR = zR[b];
  const int   dx  = dxp[0];
  const int   dy  = dyp[0];
  const float cx  = zX[b] / (float)(dx * dx);
  const float cy  = zY[b] / (float)(dy * dy);

  const float* ybase = y   + (size_t)b * (size_t)H_ * (size_t)W_;
  float*       obase = out + (size_t)b * (size_t)H_ * (size_t)W_;

  // Preload rows h0-1 (reflected), h0, h0+1 into ring slots 0,1,2.
  const int rm1 = (h0 == 0) ? 1 : (h0 - 1);
  async_b128(ybase + (size_t)rm1      * W_ + x4, &sm[0][x4]);
  async_b128(ybase + (size_t)h0       * W_ + x4, &sm[1][x4]);
  async_b128(ybase + (size_t)(h0 + 1) * W_ + x4, &sm[2][x4]);

  for (int i = 0; i < TILE; ++i) {
    // Own prefetch landed, then make all waves' LDS writes visible and ensure
    // everyone finished reading the slot we are about to recycle.
    wait_async0();
    __syncthreads();

    // Software pipeline: fetch row h+2 (reflected at the bottom edge) into the
    // slot retired last iteration while we compute row h out of LDS.
    if (i < TILE - 1) {
      int rp = h0 + i + 2;
      if (rp >= H_) rp = 2 * H_ - 2 - rp;     // reflect: H -> H-2
      async_b128(ybase + (size_t)rp * W_ + x4, &sm[(i + 3) & 3][x4]);
    }

    const int   us = i & 3, cs = (i + 1) & 3, dsl = (i + 2) & 3;
    const float* su = sm[us];                 // row h-1
    const float* sc = sm[cs];                 // row h
    const float* sd = sm[dsl];                // row h+1

    const float4 yc = *reinterpret_cast<const float4*>(&sc[x4]);
    const float4 yu = *reinterpret_cast<const float4*>(&su[x4]);
    const float4 yd = *reinterpret_cast<const float4*>(&sd[x4]);
    // In-row neighbors with width-1 reflection at the image border.
    const float yl = (x4 == 0)       ? sc[1]      : sc[x4 - 1];
    const float yr = (x4 == W_ - 4)  ? sc[W_ - 2] : sc[x4 + 4];

    // [1,-2,1] stencil along x within the lane's 4 elements.
    const float lx0 = yl   - 2.0f * yc.x + yc.y;
    const float lx1 = yc.x - 2.0f * yc.y + yc.z;
    const float lx2 = yc.y - 2.0f * yc.z + yc.w;
    const float lx3 = yc.z - 2.0f * yc.w + yr;

    float4 o;
    o.x = cx * lx0 + cy * (yu.x - 2.0f * yc.x + yd.x) + azR * yc.x * (1.0f - yc.x);
    o.y = cx * lx1 + cy * (yu.y - 2.0f * yc.y + yd.y) + azR * yc.y * (1.0f - yc.y);
    o.z = cx * lx2 + cy * (yu.z - 2.0f * yc.z + yd.z) + azR * yc.z * (1.0f - yc.z);
    o.w = cx * lx3 + cy * (yu.w - 2.0f * yc.w + yd.w) + azR * yc.w * (1.0f - yc.w);

    *reinterpret_cast<float4*>(&obase[(size_t)(h0 + i) * W_ + x4]) = o;
    // No bottom barrier needed: next iteration's wait+barrier precedes the
    // async write that recycles the slot read here.
  }
}

extern "C" void kernel_launch(void* const* d_in, const int* in_sizes, int n_in,
                              void* d_out, int out_size, void* d_ws, size_t ws_size,
                              hipStream_t stream) {
  (void)in_sizes; (void)n_in; (void)out_size; (void)d_ws; (void)ws_size;
  const float* y  = (const float*)d_in[0];
  const float* zX = (const float*)d_in[1];
  const float* zY = (const float*)d_in[2];
  const float* zR = (const float*)d_in[3];
  const int*   dx = (const int*)d_in[4];
  const int*   dy = (const int*)d_in[5];
  float* out = (float*)d_out;

  dim3 grid(H_ / TILE, B_);   // 32 x 64 = 2048 blocks, 8 wave32 per block
  ODEFunc_20710332302010_kernel<<<grid, NTHREADS, 0, stream>>>(
      y, zX, zY, zR, dx, dy, out);
}